// ReacFullGbModel_31602369364210
// MI455X (gfx1250) — compile-verified
//
#include <hip/hip_runtime.h>
#include <hip/hip_bf16.h>

// CDNA5 / gfx1250: wave32, WMMA 16x16 tiles.
typedef __attribute__((ext_vector_type(2))) float v2f;   // A/B operand of V_WMMA_F32_16X16X4_F32 (2 VGPRs)
typedef __attribute__((ext_vector_type(8))) float v8f;   // C/D operand (8 VGPRs)

#define F_FLOW 0.1f
#define V_VOL  1.0f
#define DELTA  1.0f
#define LDS_P  20   // padded LDS row stride (floats): makes 16x16 transpose stores/loads bank-conflict free

__device__ __forceinline__ float fast_tanh(float x) {
#if __has_builtin(__builtin_amdgcn_tanhf)
  return __builtin_amdgcn_tanhf(x);            // v_tanh_f32 (gfx1250 TRANS op)
#else
  float ax = __builtin_fabsf(x);
  float e  = __builtin_amdgcn_exp2f(ax * -2.8853900817779268f);   // exp(-2|x|)
  float t  = (1.0f - e) * __builtin_amdgcn_rcpf(1.0f + e);
  return __builtin_copysignf(t, x);
#endif
}

__global__ __launch_bounds__(32) void reac_rk4_wmma(
    const float* __restrict__ u,     const float* __restrict__ xz0,
    const float* __restrict__ r1_W0, const float* __restrict__ r1_b0,
    const float* __restrict__ r1_W1, const float* __restrict__ r1_b1,
    const float* __restrict__ r1_W2, const float* __restrict__ r1_b2,
    const float* __restrict__ r2_W0, const float* __restrict__ r2_b0,
    const float* __restrict__ r2_W1, const float* __restrict__ r2_b1,
    const float* __restrict__ r2_W2, const float* __restrict__ r2_b2,
    const float* __restrict__ ymean, const float* __restrict__ ystd,
    const float* __restrict__ umean, const float* __restrict__ ustd,
    float* __restrict__ yseq, float* __restrict__ xseq, int T)
{
  __shared__ __align__(16) float sh1[16 * LDS_P];
  __shared__ __align__(16) float sh2[16 * LDS_P];

  const int lane = threadIdx.x;        // one wave32 per workgroup
  const int half = lane >> 4;          // 0: lanes 0-15, 1: lanes 16-31
  const int n16  = lane & 15;
  const int row  = blockIdx.x * 16 + n16;   // batch row (mirrored across wave halves)

  // ---- hidden-layer weights, B-matrix layout for V_WMMA_F32_16X16X4_F32 ----
  // chunk c covers K rows 4c..4c+3; VGPR0={K=4c lo-half, K=4c+2 hi-half}, VGPR1={K=4c+1, K=4c+3}
  v2f w1b_r1[4], w1b_r2[4];
#pragma unroll
  for (int c = 0; c < 4; ++c) {
    int k0 = 4 * c + 2 * half;
    w1b_r1[c] = v2f{ r1_W1[k0 * 16 + n16], r1_W1[(k0 + 1) * 16 + n16] };
    w1b_r2[c] = v2f{ r2_W1[k0 * 16 + n16], r2_W1[(k0 + 1) * 16 + n16] };
  }
  // input-layer weights zero-padded to K=4
  const float zf = 0.0f;
  v2f w0b_r1 = v2f{ half ? zf : r1_W0[n16], zf };
  v2f w0b_r2 = v2f{ half ? zf : r2_W0[n16], half ? zf : r2_W0[16 + n16] };
  const float b0b_r1 = r1_b0[n16], b1b_r1 = r1_b1[n16];
  const float b0b_r2 = r2_b0[n16], b1b_r2 = r2_b1[n16];
  const float w2b_r1 = r1_W2[n16], w2b_r2 = r2_W2[n16];
  const float b2_r1  = r1_b2[0],   b2_r2  = r2_b2[0];

  // ---- normalization scalars ----
  const float Camean = ymean[0], Cbmean = ymean[1];
  const float Castd  = ystd[0],  Cbstd  = ystd[1];
  const float um = umean[0], us = ustd[0];
  const float invCa = 1.0f / Castd, invCb = 1.0f / Cbstd;
  const float rV    = F_FLOW / V_VOL;
  const float ratio = Castd * invCb;

  // ---- state (column form: lane holds its batch row; halves mirrored) ----
  float xa = xz0[row * 3 + 0];
  float xb = xz0[row * 3 + 1];
  float xc = xz0[row * 3 + 2];

  const float* __restrict__ urow = u + (size_t)row * T;
  float* __restrict__ yrow = yseq + (size_t)row * T * 2;
  float* __restrict__ xrow = xseq + (size_t)row * T * 3;

  // one RK stage: evaluates both MLPs (interleaved for ILP) + CSTR dynamics
  auto fxu = [&](float a_, float b_, float c_, float cafp,
                 float& dA, float& dB, float& dC) {
    // ---- layer 0: one zero-padded K=4 WMMA per MLP, bias preloaded in C ----
    v2f aA = v2f{ half ? zf : a_, zf };                 // r1 input: [Ca]
    v2f aB = v2f{ half ? zf : b_, half ? zf : c_ };     // r2 input: [Cb, Cc]
    v8f d1, d2;
#pragma unroll
    for (int j = 0; j < 8; ++j) { d1[j] = b0b_r1; d2[j] = b0b_r2; }
    d1 = __builtin_amdgcn_wmma_f32_16x16x4_f32(false, aA, false, w0b_r1, (short)0, d1, false, false);
    d2 = __builtin_amdgcn_wmma_f32_16x16x4_f32(false, aB, false, w0b_r2, (short)0, d2, false, false);
#pragma unroll
    for (int j = 0; j < 8; ++j) { d1[j] = fast_tanh(d1[j]); d2[j] = fast_tanh(d2[j]); }

    // ---- D-layout -> A-layout transpose through LDS (conflict-free, stride 20) ----
#pragma unroll
    for (int j = 0; j < 8; ++j) {
      sh1[(j + 8 * half) * LDS_P + n16] = d1[j];
      sh2[(j + 8 * half) * LDS_P + n16] = d2[j];
    }
    __syncthreads();                                   // single-wave WG: ~S_NOP, orders LDS
    v2f a1[4], a2[4];
#pragma unroll
    for (int c = 0; c < 4; ++c) {
      int k0 = 4 * c + 2 * half;
      const float* p1 = &sh1[n16 * LDS_P + k0];
      const float* p2 = &sh2[n16 * LDS_P + k0];
      a1[c] = v2f{ p1[0], p1[1] };                     // ds_load_b64
      a2[c] = v2f{ p2[0], p2[1] };
    }
    __syncthreads();

    // ---- hidden layer: 4 chained fp32 WMMAs per MLP ----
    v8f e1, e2;
#pragma unroll
    for (int j = 0; j < 8; ++j) { e1[j] = b1b_r1; e2[j] = b1b_r2; }
#pragma unroll
    for (int c = 0; c < 4; ++c) {
      e1 = __builtin_amdgcn_wmma_f32_16x16x4_f32(false, a1[c], false, w1b_r1[c], (short)0, e1, false, false);
      e2 = __builtin_amdgcn_wmma_f32_16x16x4_f32(false, a2[c], false, w1b_r2[c], (short)0, e2, false, false);
    }

    // ---- output layer: tanh, pre-scale by W2 (lane==column), per-row reduce via LDS ----
#pragma unroll
    for (int j = 0; j < 8; ++j) {
      e1[j] = fast_tanh(e1[j]) * w2b_r1;
      e2[j] = fast_tanh(e2[j]) * w2b_r2;
      sh1[(j + 8 * half) * LDS_P + n16] = e1[j];
      sh2[(j + 8 * half) * LDS_P + n16] = e2[j];
    }
    __syncthreads();
    float r1 = b2_r1, r2v = b2_r2;
    {
      const float4* q1 = (const float4*)&sh1[n16 * LDS_P];   // 4x ds_load_b128 (16B aligned: 80*n16)
      const float4* q2 = (const float4*)&sh2[n16 * LDS_P];
#pragma unroll
      for (int q = 0; q < 4; ++q) {
        float4 v1 = q1[q], v2 = q2[q];
        r1  += (v1.x + v1.y) + (v1.z + v1.w);
        r2v += (v2.x + v2.y) + (v2.z + v2.w);
      }
    }
    __syncthreads();

    // ---- CSTR dynamics (normalized form) ----
    float Ca_p = a_ * Castd + Camean;
    float Cb_p = b_ * Cbstd + Cbmean;
    float Cc_p = c_ * Cbstd + Cbmean;
    dA = rV * (cafp - Ca_p) * invCa - r1;
    dB = -rV * Cb_p * invCb + r1 * ratio - 3.0f * r2v;
    dC = -rV * Cc_p * invCb + r2v;
  };

  for (int t = 0; t < T; ++t) {
    // scan emits the PRE-step state
    if (lane < 16) {
      yrow[t * 2 + 0] = xa; yrow[t * 2 + 1] = xb;
      xrow[t * 3 + 0] = xa; xrow[t * 3 + 1] = xb; xrow[t * 3 + 2] = xc;
    }
    if (t + 32 < T) __builtin_prefetch(urow + t + 32, 0, 0);  // global_prefetch_b8
    float cafp = urow[t] * us + um;

    float k1a, k1b, k1c, k2a, k2b, k2c, k3a, k3b, k3c, k4a, k4b, k4c;
    fxu(xa, xb, xc, cafp, k1a, k1b, k1c);
    fxu(xa + (0.5f * DELTA) * k1a, xb + (0.5f * DELTA) * k1b, xc + (0.5f * DELTA) * k1c, cafp, k2a, k2b, k2c);
    fxu(xa + (0.5f * DELTA) * k2a, xb + (0.5f * DELTA) * k2b, xc + (0.5f * DELTA) * k2c, cafp, k3a, k3b, k3c);
    fxu(xa + DELTA * k3a, xb + DELTA * k3b, xc + DELTA * k3c, cafp, k4a, k4b, k4c);

    const float s = DELTA / 6.0f;
    xa += s * (k1a + 2.0f * k2a + 2.0f * k3a + k4a);
    xb += s * (k1b + 2.0f * k2b + 2.0f * k3b + k4b);
    xc += s * (k1c + 2.0f * k2c + 2.0f * k3c + k4c);
  }
}

extern "C" void kernel_launch(void* const* d_in, const int* in_sizes, int n_in,
                              void* d_out, int out_size, void* d_ws, size_t ws_size,
                              hipStream_t stream) {
  const float* u   = (const float*)d_in[0];
  const float* xz0 = (const float*)d_in[1];
  const int B = in_sizes[1] / 3;        // xz0 is (B,3)
  const int T = in_sizes[0] / B;        // u is (B,T,1)
  float* yseq = (float*)d_out;
  float* xseq = yseq + (size_t)B * T * 2;

  dim3 grid(B / 16), block(32);
  reac_rk4_wmma<<<grid, block, 0, stream>>>(
      u, xz0,
      (const float*)d_in[2],  (const float*)d_in[3],  (const float*)d_in[4],  (const float*)d_in[5],
      (const float*)d_in[6],  (const float*)d_in[7],  (const float*)d_in[8],  (const float*)d_in[9],
      (const float*)d_in[10], (const float*)d_in[11], (const float*)d_in[12], (const float*)d_in[13],
      (const float*)d_in[14], (const float*)d_in[15], (const float*)d_in[16], (const float*)d_in[17],
      yseq, xseq, T);
}